// StructuralDistillationLoss_87230785781773
// MI455X (gfx1250) — compile-verified
//
#include <hip/hip_runtime.h>

typedef float v2f __attribute__((ext_vector_type(2)));
typedef float v8f __attribute__((ext_vector_type(8)));

#define IMG_H 128
#define IMG_W 128
#define NCH   64
#define NCP   32            // channel pairs
#define TW    32            // tile width  (output pixels)
#define TH    8             // tile height (output pixels)
#define XP    34            // tile cols incl. 1-pixel halo
#define YT    10            // tile rows incl. 1-pixel halo
#define PIX   (XP * YT)     // 340 pixels per tile (incl. halo)
#define PIXP  368           // padded pixel stride per channel-pair (v2f units);
                            // 2*PIXP = 736 ≡ 32 (mod 64) -> b64 gathers hit
                            // complementary bank halves for lanes 0-15 / 16-31
#define CPF   (2 * PIXP)    // 736 floats per channel-pair slab
#define FEATF (NCP * CPF)   // 23552 floats per feature tile
#define NWG   512           // 8 batches * 4 x-tiles * 16 y-tiles
#define EPSF  1e-12f

extern __shared__ float smem[];
// smem layout: sF[FEATF] | tF[FEATF] | sInv[PIX] | tInv[PIX] | red[256]

__global__ void affinity_mse_kernel(const float* __restrict__ S,
                                    const float* __restrict__ T,
                                    float* __restrict__ partial) {
    float* sF   = smem;
    float* tF   = smem + FEATF;
    float* sInv = smem + 2 * FEATF;
    float* tInv = sInv + PIX;
    float* red  = tInv + PIX;
    v2f* sF2 = (v2f*)sF;   // v2f index: cp*PIXP + pixel
    v2f* tF2 = (v2f*)tF;

    const int tid = threadIdx.x;
    const int wg  = blockIdx.x;
    const int tx  = wg & 3;          // 4 tiles in x
    const int ty  = (wg >> 2) & 15;  // 16 tiles in y
    const int b   = wg >> 6;         // 8 batches

    const int gx0 = tx * TW - 1;
    const int gy0 = ty * TH - 1;
    const long long base = (long long)b * NCH * IMG_H * IMG_W;

    // ---- Phase 1: stream tile + halo into LDS (pair-interleaved), zero halo ----
    for (int e = tid; e < NCH * PIX; e += 256) {
        int c   = e / PIX;
        int p   = e - c * PIX;       // p = ry*XP + rx
        int ry  = p / XP;
        int rx  = p - ry * XP;
        int gy = gy0 + ry, gx = gx0 + rx;
        float vs = 0.0f, vt = 0.0f;
        if (gy >= 0 && gy < IMG_H && gx >= 0 && gx < IMG_W) {
            long long idx = base + ((long long)c * IMG_H + gy) * IMG_W + gx;
            vs = S[idx];
            vt = T[idx];
        }
        int li = (c >> 1) * CPF + p * 2 + (c & 1);
        sF[li] = vs;
        tF[li] = vt;
    }
    __syncthreads();

    // ---- Phase 2: per-pixel inverse L2 norms (F.normalize eps semantics) ----
    for (int p = tid; p < PIX; p += 256) {
        float ss = 0.0f, st = 0.0f;
        for (int cp = 0; cp < NCP; ++cp) {
            v2f a = sF2[cp * PIXP + p]; ss += a.x * a.x + a.y * a.y;
            v2f d = tF2[cp * PIXP + p]; st += d.x * d.x + d.y * d.y;
        }
        sInv[p] = 1.0f / fmaxf(sqrtf(ss), EPSF);
        tInv[p] = 1.0f / fmaxf(sqrtf(st), EPSF);
    }
    __syncthreads();

    float acc = 0.0f;

    // ---- Phase 3: WMMA Gram jobs ----
    // 16 jobs = y(8) x segment(2); each job computes the three neighbor-row
    // Grams (di = -1,0,+1) for student and teacher, sharing the A matrix.
    // Wave-uniform loop => EXEC all ones as WMMA requires.
    const int lane = tid & 31;
    const int wave = tid >> 5;
    const int half = lane >> 4;   // K-half (A/B layout) == M-half (C layout)
    const int ln   = lane & 15;   // A: row m; B/C: col n

    for (int j = wave; j < 16; j += 8) {
        int seg = j & 1;
        int y   = j >> 1;
        int ryP = y + 1;           // LDS row of output pixels
        int x   = 1 + seg * 16;    // LDS col base of the 16-pixel segment
        int pcol = x + ln;

        // Per K-step, lane holds channels {4*kb+2*half, +1}: cp = 2*kb + half.
        int ja  = half * PIXP + (ryP * XP + pcol);        // A: row ryP
        int jb  = half * PIXP + ((ryP - 1) * XP + pcol);  // B: rows ryP-1..ryP+1

        v8f cs0 = {}, cs1 = {}, cs2 = {};  // student Grams, di=-1,0,+1
        v8f ct0 = {}, ct1 = {}, ct2 = {};  // teacher Grams
#pragma unroll
        for (int kb = 0; kb < 16; ++kb) {   // K = 64 via 16 chained K=4 steps
            v2f aS = sF2[ja];
            v2f aT = tF2[ja];
            v2f b0S = sF2[jb], b1S = sF2[jb + XP], b2S = sF2[jb + 2 * XP];
            v2f b0T = tF2[jb], b1T = tF2[jb + XP], b2T = tF2[jb + 2 * XP];
            cs0 = __builtin_amdgcn_wmma_f32_16x16x4_f32(false, aS, false, b0S,
                                                        (short)0, cs0, false, false);
            cs1 = __builtin_amdgcn_wmma_f32_16x16x4_f32(false, aS, false, b1S,
                                                        (short)0, cs1, false, false);
            cs2 = __builtin_amdgcn_wmma_f32_16x16x4_f32(false, aS, false, b2S,
                                                        (short)0, cs2, false, false);
            ct0 = __builtin_amdgcn_wmma_f32_16x16x4_f32(false, aT, false, b0T,
                                                        (short)0, ct0, false, false);
            ct1 = __builtin_amdgcn_wmma_f32_16x16x4_f32(false, aT, false, b1T,
                                                        (short)0, ct1, false, false);
            ct2 = __builtin_amdgcn_wmma_f32_16x16x4_f32(false, aT, false, b2T,
                                                        (short)0, ct2, false, false);
            ja += 2 * PIXP;
            jb += 2 * PIXP;
        }

        // Consume the 3 diagonals |n-m| <= 1 of each Gram; every needed entry
        // lives in this lane's own C slice (lane = col n, VGPR r = row m).
#pragma unroll
        for (int di2 = 0; di2 < 3; ++di2) {
            v8f cs = (di2 == 0) ? cs0 : (di2 == 1) ? cs1 : cs2;
            v8f ct = (di2 == 0) ? ct0 : (di2 == 1) ? ct1 : ct2;
            int ryQ = ryP - 1 + di2;
            int qn = ryQ * XP + pcol;      // neighbor pixel (n = ln)
            float invSq = sInv[qn], invTq = tInv[qn];
#pragma unroll
            for (int r = 0; r < 8; ++r) {
                int m = r + 8 * half;
                int d = ln - m;
                if (d >= -1 && d <= 1) {
                    int pm = ryP * XP + x + m;
                    float sa = cs[r] * sInv[pm] * invSq;
                    float ta = ct[r] * tInv[pm] * invTq;
                    float df = sa - ta;
                    acc += df * df;
                }
            }
        }
    }

    // ---- Phase 4: VALU fixup for the 2 cross-segment pairs per (y,di,seg) ----
    if (tid < 96) {
        int t    = tid;
        int side = t & 1;
        int seg  = (t >> 1) & 1;
        int di   = ((t >> 2) % 3) - 1;
        int y    = (t >> 2) / 3;
        int ryP = y + 1, ryQ = y + 1 + di;
        int xP = 1 + seg * 16 + side * 15;
        int xQ = xP + (side ? 1 : -1);
        int pi = ryP * XP + xP;
        int qi = ryQ * XP + xQ;
        float ds = 0.0f, dt = 0.0f;
        for (int cp = 0; cp < NCP; ++cp) {
            v2f sp = sF2[cp * PIXP + pi], sq = sF2[cp * PIXP + qi];
            v2f tp = tF2[cp * PIXP + pi], tq = tF2[cp * PIXP + qi];
            ds += sp.x * sq.x + sp.y * sq.y;
            dt += tp.x * tq.x + tp.y * tq.y;
        }
        float sa = ds * sInv[pi] * sInv[qi];
        float ta = dt * tInv[pi] * tInv[qi];
        float df = sa - ta;
        acc += df * df;
    }

    // ---- Phase 5: block reduction -> per-WG partial ----
    __syncthreads();           // done reading feature LDS
    red[tid] = acc;
    __syncthreads();
    for (int s = 128; s > 0; s >>= 1) {
        if (tid < s) red[tid] += red[tid + s];
        __syncthreads();
    }
    if (tid == 0) partial[wg] = red[0];
}

__global__ void final_reduce_kernel(const float* __restrict__ partial,
                                    float* __restrict__ out) {
    __shared__ float red[256];
    int tid = threadIdx.x;
    red[tid] = partial[tid] + partial[tid + 256];
    __syncthreads();
    for (int s = 128; s > 0; s >>= 1) {
        if (tid < s) red[tid] += red[tid + s];
        __syncthreads();
    }
    if (tid == 0)
        out[0] = red[0] * (1.0f / (8.0f * 9.0f * 128.0f * 128.0f));
}

extern "C" void kernel_launch(void* const* d_in, const int* in_sizes, int n_in,
                              void* d_out, int out_size, void* d_ws, size_t ws_size,
                              hipStream_t stream) {
    const float* S = (const float*)d_in[0];   // student_feat (8,64,128,128) f32
    const float* T = (const float*)d_in[1];   // teacher_feat
    float* partial = (float*)d_ws;            // 512 floats of scratch

    size_t shmem = (size_t)(2 * FEATF + 2 * PIX + 256) * sizeof(float); // ~192 KB (< 320 KB WGP LDS)

    affinity_mse_kernel<<<NWG, 256, shmem, stream>>>(S, T, partial);
    final_reduce_kernel<<<1, 256, 0, stream>>>(partial, (float*)d_out);
}